// HHNet_No_Gating_74500502717131
// MI455X (gfx1250) — compile-verified
//
#include <hip/hip_runtime.h>

// ---------------------------------------------------------------------------
// HH spiking net for MI455X (gfx1250, wave32).
// Pipeline:  z1 = batch @ W1^T   (WMMA f32 16x16x4, multi-n-tile per wave)
//            T1 = sigmoid-of-HH-scan(z1)   (sequential over T, 3200 channels)
//            z2 = T1 @ W2^T      (WMMA f32 16x16x4)
//            out = sigmoid-of-HH-scan(z2)  (320 channels)
//
// Data-movement plan: batch (100 MB) is streamed exactly ONCE (each wave owns
// one 16-row m-tile and computes all n-tiles), W1/W2 are L2-resident.
// ---------------------------------------------------------------------------

typedef float v2f __attribute__((ext_vector_type(2)));
typedef float v8f __attribute__((ext_vector_type(8)));

// Problem shape (fixed by setup_inputs in the reference)
#define BB   32
#define TT   1000
#define NI   784          // 28*28
#define NH   100
#define NO   10
#define MROWS (BB * TT)   // 32000

// HH constants
#define C_GNA 40.0f
#define C_GK  35.0f
#define C_GL  0.1f
#define C_ENA 55.0f
#define C_EK  (-77.0f)
#define C_EL  (-65.0f)
#define C_IAPP 0.5f
#define C_VT  20.0f
#define C_KP  3.0f
#define C_DT  0.05f

__device__ __forceinline__ float frcp(float x) { return __builtin_amdgcn_rcpf(x); }
__device__ __forceinline__ float fexp(float x) { return __expf(x); }
__device__ __forceinline__ float fsigmoid(float x) { return frcp(1.0f + fexp(-x)); }

// ---------------------------------------------------------------------------
// WMMA f32 GEMM:  Z[M x N] = A[M x K] * W[N x K]^T
// One 16-row m-tile per wave32; the wave keeps NTILES 16x16 f32 accumulators
// (one per n-tile) so each A fragment is loaded once and fed to NTILES
// independent V_WMMA_F32_16X16X4_F32 ops (no C-RAW between consecutive WMMAs).
//
// A-fragment (16x4 f32, ISA 7.12.2): lane l holds row m = l&15,
//   K pair {k0 + 2*(l>>4), +1}  -> one contiguous float2 load per lane; over
//   the K loop lanes l and l+16 jointly stream row m contiguously.
// B-fragment (4x16 f32): mirror of A layout: lane l holds col n = l&15,
//   K rows {k0 + 2*(l>>4), +1} -> contiguous float2 from W[n,:].
// D layout: lane l holds N = l&15, VGPR v holds M = v + 8*(l>>4).
// EXEC is full for every wave that reaches the WMMA loop (wave-uniform guard).
// ---------------------------------------------------------------------------
template <int K, int N, int NTILES>
__global__ __launch_bounds__(256) void gemm_wmma_f32(const float* __restrict__ A,
                                                     const float* __restrict__ W,
                                                     float* __restrict__ Z,
                                                     int num_mtiles) {
    const int lane = threadIdx.x & 31;
    const int wave = threadIdx.x >> 5;
    const int mt   = blockIdx.x * 8 + wave;
    if (mt >= num_mtiles) return;                // wave-uniform

    const int M0    = mt * 16;
    const int r     = lane & 15;
    const int khalf = (lane >> 4) << 1;          // 0 or 2

    // A: row (M0 + r), cols k0+khalf .. +1
    const float* aptr = A + (size_t)(M0 + r) * K + khalf;

    // One B base pointer per n-tile (row clamped; store is masked for pad cols)
    const float* bptr[NTILES];
#pragma unroll
    for (int j = 0; j < NTILES; ++j) {
        int n = j * 16 + r;
        if (n > N - 1) n = N - 1;
        bptr[j] = W + (size_t)n * K + khalf;
    }

    v8f acc[NTILES];
#pragma unroll
    for (int j = 0; j < NTILES; ++j) {
        v8f zero = {0.f, 0.f, 0.f, 0.f, 0.f, 0.f, 0.f, 0.f};
        acc[j] = zero;
    }

#pragma unroll 2
    for (int k0 = 0; k0 < K; k0 += 4) {
        const v2f av = *(const v2f*)(aptr + k0);   // shared across all n-tiles
#pragma unroll
        for (int j = 0; j < NTILES; ++j) {
            v2f bv = *(const v2f*)(bptr[j] + k0);
            acc[j] = __builtin_amdgcn_wmma_f32_16x16x4_f32(
                /*neg_a=*/false, av, /*neg_b=*/false, bv,
                /*c_mod=*/(short)0, acc[j], /*reuse_a=*/false, /*reuse_b=*/false);
        }
    }

    const int gn0   = lane & 15;
    const int mbase = M0 + ((lane >> 4) << 3);
#pragma unroll
    for (int j = 0; j < NTILES; ++j) {
        const int gn = j * 16 + gn0;
        if (gn < N) {
#pragma unroll
            for (int v = 0; v < 8; ++v) {
                Z[(size_t)(mbase + v) * N + gn] = acc[j][v];
            }
        }
    }
}

// ---------------------------------------------------------------------------
// Hodgkin-Huxley scan: one thread per (batch, feature) channel, sequential
// over T. State (V, m, n, h) lives in registers; emits sigmoid((V-VT)/KP).
// Crank-Nicolson gate update; singular-point patches at V==25 / V==-35.
// 4x v_exp_f32 + shared v_rcp_f32 per step (latency chain is the bottleneck).
// ---------------------------------------------------------------------------
__device__ __forceinline__ float gate_update(float a, float b, float g) {
    float s = C_DT * 0.5f * (a + b);
    return (a * C_DT + (1.0f - s) * g) * frcp(1.0f + s);
}

__global__ void hh_scan(const float* __restrict__ Zin, float* __restrict__ Tout,
                        int F) {
    const int tid = blockIdx.x * blockDim.x + threadIdx.x;
    const int channels = BB * F;
    if (tid >= channels) return;
    const int b = tid / F;
    const int f = tid - b * F;
    const size_t base = (size_t)b * TT * F + f;

    float V = -70.0f, m = 0.0f, n = 0.0f, h = 1.0f;

    for (int t = 0; t < TT; ++t) {
        Tout[base + (size_t)t * F] = fsigmoid((V - C_VT) * (1.0f / C_KP));
        if (t == TT - 1) break;

        const float z = Zin[base + (size_t)t * F];

        // implicit voltage update
        const float pow1 = C_GNA * m * m * m * h;
        const float n2   = n * n;
        const float pow2 = C_GK * n2 * n2;
        const float G    = C_DT * 0.5f * (pow1 + pow2 + C_GL);
        const float E    = pow1 * C_ENA + pow2 * C_EK + C_GL * C_EL;
        V = (V * (1.0f - G) + C_DT * (E + C_IAPP + z)) * frcp(1.0f + G);

        // channel rates (4 exps; shared reciprocals for a/b pairs)
        const float en = fexp((V - 25.0f) * (1.0f / 9.0f));
        const float em = fexp((V + 35.0f) * (1.0f / 9.0f));
        const float rn = frcp(en - 1.0f);
        const float rm = frcp(em - 1.0f);
        float aN = 0.02f  * (V - 25.0f) * en * rn;   // 0.02(V-25)/(1-exp((25-V)/9))
        float bN = 0.002f * (V - 25.0f) * rn;        // -0.002(V-25)/(1-exp((V-25)/9))
        float aM = 0.182f * (V + 35.0f) * em * rm;
        float bM = 0.124f * (V + 35.0f) * rm;
        const float aH = 0.25f * fexp((-V - 90.0f) * (1.0f / 12.0f));
        const float bH = 0.25f * fexp((V + 34.0f) * (1.0f / 12.0f));

        if (V == 25.0f)  { aN = 0.18f;  bN = 0.018f; }
        if (V == -35.0f) { aM = 1.638f; bM = 1.116f; }

        m = gate_update(aM, bM, m);
        n = gate_update(aN, bN, n);
        h = gate_update(aH, bH, h);
    }
}

// ---------------------------------------------------------------------------
extern "C" void kernel_launch(void* const* d_in, const int* in_sizes, int n_in,
                              void* d_out, int out_size, void* d_ws, size_t ws_size,
                              hipStream_t stream) {
    const float* batch = (const float*)d_in[0]; // (32,1000,784)
    const float* W1    = (const float*)d_in[1]; // (100,784)
    const float* W2    = (const float*)d_in[2]; // (10,100)
    float*       out   = (float*)d_out;         // (32,1000,10)

    float* z1 = (float*)d_ws;                   // 32000*100 f32 = 12.8 MB
    float* t1 = z1 + (size_t)MROWS * NH;        // 12.8 MB
    float* z2 = t1 + (size_t)MROWS * NH;        // 32000*10 f32 = 1.28 MB

    const int mtiles = MROWS / 16;              // 2000 = 250 blocks * 8 waves

    // GEMM1: each wave owns one m-tile, computes all 7 n-tiles (N=100 -> 112)
    gemm_wmma_f32<NI, NH, 7><<<mtiles / 8, 256, 0, stream>>>(batch, W1, z1, mtiles);

    // HH scan layer 1: 3200 channels
    hh_scan<<<(BB * NH + 255) / 256, 256, 0, stream>>>(z1, t1, NH);

    // GEMM2: one n-tile (N=10 -> pad to 16)
    gemm_wmma_f32<NH, NO, 1><<<mtiles / 8, 256, 0, stream>>>(t1, W2, z2, mtiles);

    // HH scan layer 2: 320 channels, write sigmoid output directly
    hh_scan<<<(BB * NO + 63) / 64, 64, 0, stream>>>(z2, out, NO);
}